// Uni3DViewTrans_49718541418847
// MI455X (gfx1250) — compile-verified
//
#include <hip/hip_runtime.h>
#include <hip/hip_bf16.h>

#define VXD 80
#define VYD 80
#define VZD 5
#define QN  (VXD*VYD*VZD)   // 32000
#define CCH 128
#define NCAM 6
#define IMG_HF 928.0f
#define IMG_WF 1600.0f
#define DDIM 64
#define DH 116
#define DW 200

typedef float v2f __attribute__((ext_vector_type(2)));
typedef float v8f __attribute__((ext_vector_type(8)));

// ---------------- Stage 1: projection + trilinear depth sample + mask ----------------
__global__ __launch_bounds__(256) void k_project(const float* __restrict__ depth,
                                                 const float* __restrict__ l2i,
                                                 float* __restrict__ gxv,
                                                 float* __restrict__ gyv,
                                                 float* __restrict__ dmv) {
  int idx = blockIdx.x * 256 + threadIdx.x;
  if (idx >= NCAM * QN) return;
  int n = idx / QN;
  int q = idx - n * QN;
  int ix = q / (VYD * VZD);
  int r  = q - ix * (VYD * VZD);
  int iy = r / VZD;
  int iz = r - iy * VZD;
  float px = (float)ix * (108.0f / 79.0f) - 54.0f;
  float py = (float)iy * (108.0f / 79.0f) - 54.0f;
  float pz = (float)iz * (8.0f / 4.0f) - 5.0f;
  const float* M = l2i + n * 16;
  float p0 = M[0]*px + M[1]*py + M[2]*pz  + M[3];
  float p1 = M[4]*px + M[5]*py + M[6]*pz  + M[7];
  float p2 = M[8]*px + M[9]*py + M[10]*pz + M[11];
  bool valid = p2 > 1e-5f;
  float zc = fmaxf(p2, 1e-5f);
  float gx = (p0 / (zc * IMG_WF) - 0.5f) * 2.0f;
  float gy = (p1 / (zc * IMG_HF) - 0.5f) * 2.0f;
  float gz = (p2 * (1.0f / (float)DDIM) - 0.5f) * 2.0f;
  bool mask = valid && gx > -1.0f && gx < 1.0f && gy > -1.0f && gy < 1.0f
                    && gz > -1.0f && gz < 1.0f;
  float xx = (gx + 1.0f) * (DW * 0.5f) - 0.5f;
  float yy = (gy + 1.0f) * (DH * 0.5f) - 0.5f;
  float zz = (gz + 1.0f) * (DDIM * 0.5f) - 0.5f;
  float xf = floorf(xx), yf = floorf(yy), zf = floorf(zz);
  float wx = xx - xf, wy = yy - yf, wz = zz - zf;
  int x0 = (int)xf, y0 = (int)yf, z0 = (int)zf;
  const float* V = depth + (size_t)n * DDIM * DH * DW;
  float sd = 0.0f;
  #pragma unroll
  for (int dz = 0; dz < 2; ++dz)
    #pragma unroll
    for (int dy = 0; dy < 2; ++dy)
      #pragma unroll
      for (int dx = 0; dx < 2; ++dx) {
        int cx = x0 + dx, cy = y0 + dy, cz = z0 + dz;
        if (cx >= 0 && cx < DW && cy >= 0 && cy < DH && cz >= 0 && cz < DDIM) {
          float wgt = (dx ? wx : 1.0f - wx) * (dy ? wy : 1.0f - wy) * (dz ? wz : 1.0f - wz);
          sd += wgt * V[((size_t)cz * DH + cy) * DW + cx];
        }
      }
  gxv[idx] = gx;
  gyv[idx] = gy;
  dmv[idx] = mask ? sd : 0.0f;
}

// ---------------- Stage 2: multi-level bilinear gather, weighted camera sum ----------------
__global__ __launch_bounds__(128) void k_sample(const float* __restrict__ f0,
                                                const float* __restrict__ f1,
                                                const float* __restrict__ f2,
                                                const float* __restrict__ f3,
                                                const float* __restrict__ gxv,
                                                const float* __restrict__ gyv,
                                                const float* __restrict__ dmv,
                                                float* __restrict__ vox) {
  int q = blockIdx.x;
  int c = threadIdx.x;
  const float* fs[4] = {f0, f1, f2, f3};
  const int HS[4] = {116, 58, 29, 15};
  const int WS[4] = {200, 100, 50, 25};
  float acc = 0.0f;
  for (int n = 0; n < NCAM; ++n) {
    float dm = dmv[n * QN + q];
    if (dm == 0.0f) continue;   // masked camera: skip all 4 levels (uniform branch)
    float gx = gxv[n * QN + q];
    float gy = gyv[n * QN + q];
    #pragma unroll
    for (int l = 0; l < 4; ++l) {
      int H = HS[l], W = WS[l];
      const float* f = fs[l] + (size_t)n * CCH * H * W;
      float xx = (gx + 1.0f) * (W * 0.5f) - 0.5f;
      float yy = (gy + 1.0f) * (H * 0.5f) - 0.5f;
      float xf = floorf(xx), yf = floorf(yy);
      float wx = xx - xf, wy = yy - yf;
      int x0 = (int)xf, y0 = (int)yf;
      float s = 0.0f;
      #pragma unroll
      for (int dy = 0; dy < 2; ++dy)
        #pragma unroll
        for (int dx = 0; dx < 2; ++dx) {
          int cx = x0 + dx, cy = y0 + dy;
          if (cx >= 0 && cx < W && cy >= 0 && cy < H) {
            float wgt = (dx ? wx : 1.0f - wx) * (dy ? wy : 1.0f - wy);
            s += wgt * f[((size_t)c * H + cy) * W + cx];   // L2-resident gather
          }
        }
      acc += dm * s;
    }
  }
  int ix = q / (VYD * VZD);
  int r  = q - ix * (VYD * VZD);
  int iy = r / VZD;
  int iz = r - iy * VZD;
  vox[(((size_t)c * VZD + iz) * VYD + iy) * VXD + ix] = acc;
}

// ---------------- Weight repack into WMMA A-fragment order ----------------
// packed[t][c4][co_tile][lane][2]  with  co = co_tile*16 + (lane&15),
// k = c4*4 + (lane>>4)*2 + {0,1}, value = w[co][ci=k][ky][kx], t = ky*3+kx.
// A-fragment load becomes one coalesced global_load_b64 per lane.
__global__ __launch_bounds__(256) void k_repack(const float* __restrict__ w,
                                                float* __restrict__ pw) {
  int idx = blockIdx.x * 256 + threadIdx.x;   // over 9*32*8*32 = 73728 entries
  if (idx >= 9 * 32 * 8 * 32) return;
  int lane = idx & 31;
  int rest = idx >> 5;
  int cot  = rest & 7;
  rest >>= 3;
  int c4 = rest & 31;
  int t  = rest >> 5;
  int co = cot * 16 + (lane & 15);
  int k0 = c4 * 4 + (lane >> 4) * 2;
  pw[(size_t)idx * 2 + 0] = w[((size_t)co * CCH + k0 + 0) * 9 + t];
  pw[(size_t)idx * 2 + 1] = w[((size_t)co * CCH + k0 + 1) * 9 + t];
}

// ---------------- Stage 3: (1,3,3) conv as implicit GEMM via V_WMMA_F32_16X16X4_F32 ----------
// Block: 256 threads = 8 waves; computes 128 c_out x (16x X 2y) at fixed z.
// Each wave: one 16-c_out stripe, two 16x16 N-tiles sharing every A fragment.
// LDS tile is channel-pair interleaved: [(ci>>1)][row 0..3][col 0..17][ci&1]
// so every B fragment (ci=k0, ci=k0+1 at same spatial pos, k0 even) is one
// aligned 8-byte pair -> single ds_load_b64, no register shuffles.
// Per K-step: 1 coalesced global b64 (A) + 2 ds_load_b64 (B0,B1) + 2 WMMAs.
__global__ __launch_bounds__(256) void k_conv3x3(const float* __restrict__ in,
                                                 const float* __restrict__ pw,
                                                 const float* __restrict__ bias,
                                                 float* __restrict__ out) {
  __shared__ float smem[CCH * 72];   // 64 ci-pairs * 4 rows * 18 cols * 2, 36.9 KB
  int bid = blockIdx.x;              // grid = VZD * (VYD/2) * (VXD/16) = 1000
  int z   = bid / ((VYD / 2) * 5);
  int rem = bid - z * ((VYD / 2) * 5);
  int yp  = rem / 5;
  int x0  = (rem - yp * 5) * 16;
  int y0  = yp * 2;                  // output rows y0, y0+1

  // stage input patch rows y0-1 .. y0+2 (zero-padded), writing interleaved layout
  for (int e = threadIdx.x; e < CCH * 72; e += 256) {
    int ci  = e / 72;
    int rr  = (e % 72) / 18;
    int col = e % 18;
    int yy = y0 + rr - 1;
    int xx = x0 + col - 1;
    float v = 0.0f;
    if (yy >= 0 && yy < VYD && xx >= 0 && xx < VXD)
      v = in[(((size_t)ci * VZD + z) * VYD + yy) * VXD + xx];
    smem[(((ci >> 1) * 72) + rr * 18 + col) * 2 + (ci & 1)] = v;
  }
  __syncthreads();

  int wave = threadIdx.x >> 5;
  int lane = threadIdx.x & 31;
  int nn   = lane & 15;     // N (spatial x) index; also M row for A fragment
  int hi   = lane >> 4;     // lane-half -> K split (K = v + 2*hi)

  v8f acc0 = {};            // output row y0
  v8f acc1 = {};            // output row y0+1
  for (int t = 0; t < 9; ++t) {
    int ky = t / 3, kx = t % 3;
    const float* ap  = pw + ((((size_t)t * 32) * 8 + wave) * 32 + lane) * 2;
    // base of this lane's fragment pair: ci-pair index (hi) + spatial (ky..,kx+nn)
    const float* sb0 = smem + hi * 144 + (ky * 18 + kx + nn) * 2;
    const float* sb1 = sb0 + 36;      // next output row (ky+1)
    #pragma unroll 4
    for (int c4 = 0; c4 < CCH / 4; ++c4) {
      v2f a  = *(const v2f*)(ap  + (size_t)c4 * 8 * 32 * 2);   // coalesced b64
      v2f b0 = *(const v2f*)(sb0 + (size_t)c4 * 288);          // ds_load_b64
      v2f b1 = *(const v2f*)(sb1 + (size_t)c4 * 288);          // ds_load_b64
      acc0 = __builtin_amdgcn_wmma_f32_16x16x4_f32(false, a, false, b0,
                                                   (short)0, acc0, false, false);
      acc1 = __builtin_amdgcn_wmma_f32_16x16x4_f32(false, a, false, b1,
                                                   (short)0, acc1, false, false);
    }
  }

  // D layout: VGPR v -> row M = v + 8*hi, col N = lane&15
  int xo = x0 + nn;
  #pragma unroll
  for (int v = 0; v < 8; ++v) {
    int coo = wave * 16 + hi * 8 + v;
    float bb = bias[coo];
    size_t base = ((size_t)coo * VZD + z) * VYD;
    out[(base + y0 + 0) * VXD + xo] = acc0[v] + bb;
    out[(base + y0 + 1) * VXD + xo] = acc1[v] + bb;
  }
}

// ---------------- Stage 4: per-channel mean/var + normalize + ReLU ----------------
__global__ __launch_bounds__(256) void k_stats(const float* __restrict__ x,
                                               float* __restrict__ stats) {
  __shared__ float s1[256];
  __shared__ float s2[256];
  int c = blockIdx.x;
  const float* p = x + (size_t)c * QN;
  float a = 0.0f, b = 0.0f;
  for (int i = threadIdx.x; i < QN; i += 256) {
    float v = p[i];
    a += v;
    b += v * v;
  }
  s1[threadIdx.x] = a;
  s2[threadIdx.x] = b;
  __syncthreads();
  for (int o = 128; o > 0; o >>= 1) {
    if (threadIdx.x < o) {
      s1[threadIdx.x] += s1[threadIdx.x + o];
      s2[threadIdx.x] += s2[threadIdx.x + o];
    }
    __syncthreads();
  }
  if (threadIdx.x == 0) {
    float mu = s1[0] * (1.0f / (float)QN);
    stats[c] = mu;
    stats[CCH + c] = s2[0] * (1.0f / (float)QN) - mu * mu;
  }
}

__global__ __launch_bounds__(256) void k_norm(const float* __restrict__ in,
                                              float* __restrict__ out,
                                              const float* __restrict__ stats,
                                              const float* __restrict__ g,
                                              const float* __restrict__ be) {
  int idx = blockIdx.x * 256 + threadIdx.x;
  if (idx >= CCH * QN) return;
  int c = idx / QN;
  float mu  = stats[c];
  float var = stats[CCH + c];
  float v = (in[idx] - mu) * rsqrtf(var + 1e-5f) * g[c] + be[c];
  out[idx] = fmaxf(v, 0.0f);
}

// ---------------- Launch ----------------
extern "C" void kernel_launch(void* const* d_in, const int* in_sizes, int n_in,
                              void* d_out, int out_size, void* d_ws, size_t ws_size,
                              hipStream_t stream) {
  const float* f0    = (const float*)d_in[0];
  const float* f1    = (const float*)d_in[1];
  const float* f2    = (const float*)d_in[2];
  const float* f3    = (const float*)d_in[3];
  const float* depth = (const float*)d_in[4];
  const float* l2i   = (const float*)d_in[5];
  const float* w1  = (const float*)d_in[6];
  const float* b1  = (const float*)d_in[7];
  const float* g1  = (const float*)d_in[8];
  const float* be1 = (const float*)d_in[9];
  const float* w2  = (const float*)d_in[10];
  const float* b2  = (const float*)d_in[11];
  const float* g2  = (const float*)d_in[12];
  const float* be2 = (const float*)d_in[13];
  const float* w3  = (const float*)d_in[14];
  const float* b3  = (const float*)d_in[15];
  const float* g3  = (const float*)d_in[16];
  const float* be3 = (const float*)d_in[17];
  float* out = (float*)d_out;

  const size_t NQ = (size_t)NCAM * QN;
  const size_t PWN = (size_t)9 * 32 * 8 * 32 * 2;   // packed weight floats per layer
  char* ws = (char*)d_ws;
  float* gxv = (float*)ws;
  float* gyv = gxv + NQ;
  float* dmv = gyv + NQ;
  size_t off = ((3 * NQ * sizeof(float)) + 255) & ~(size_t)255;
  float* bufA  = (float*)(ws + off);
  float* bufB  = bufA + (size_t)CCH * QN;
  float* stats = bufB + (size_t)CCH * QN;   // 2*CCH floats
  float* pw1   = stats + 2 * CCH + 64;      // keep 256B-ish alignment slack
  float* pw2   = pw1 + PWN;
  float* pw3   = pw2 + PWN;

  const int convBlocks   = VZD * (VYD / 2) * (VXD / 16);  // 1000
  const int normBlocks   = (CCH * QN + 255) / 256;
  const int repackBlocks = (9 * 32 * 8 * 32 + 255) / 256; // 288

  // independent prep work
  k_repack<<<repackBlocks, 256, 0, stream>>>(w1, pw1);
  k_repack<<<repackBlocks, 256, 0, stream>>>(w2, pw2);
  k_repack<<<repackBlocks, 256, 0, stream>>>(w3, pw3);
  k_project<<<(int)((NQ + 255) / 256), 256, 0, stream>>>(depth, l2i, gxv, gyv, dmv);
  k_sample<<<QN, 128, 0, stream>>>(f0, f1, f2, f3, gxv, gyv, dmv, bufA);

  // layer 1
  k_conv3x3<<<convBlocks, 256, 0, stream>>>(bufA, pw1, b1, bufB);
  k_stats<<<CCH, 256, 0, stream>>>(bufB, stats);
  k_norm<<<normBlocks, 256, 0, stream>>>(bufB, bufB, stats, g1, be1);
  // layer 2
  k_conv3x3<<<convBlocks, 256, 0, stream>>>(bufB, pw2, b2, bufA);
  k_stats<<<CCH, 256, 0, stream>>>(bufA, stats);
  k_norm<<<normBlocks, 256, 0, stream>>>(bufA, bufA, stats, g2, be2);
  // layer 3 -> final output
  k_conv3x3<<<convBlocks, 256, 0, stream>>>(bufA, pw3, b3, bufB);
  k_stats<<<CCH, 256, 0, stream>>>(bufB, stats);
  k_norm<<<normBlocks, 256, 0, stream>>>(bufB, out, stats, g3, be3);
}